// EGCN_20538533609947
// MI455X (gfx1250) — compile-verified
//
#include <hip/hip_runtime.h>
#include <cstddef>

#define SLOPE (11.0f / 48.0f)

typedef float v2f __attribute__((ext_vector_type(2)));
typedef float v8f __attribute__((ext_vector_type(8)));

constexpr int TSTEPS = 6;
constexpr int NF = 256;   // feature/hidden width (K and N of all big GEMMs)

__device__ __forceinline__ float rrelu_f(float x) { return x >= 0.f ? x : SLOPE * x; }

// ----------------------------- zero fill (ints) -----------------------------
__global__ void zero_i32(int* __restrict__ p, long n) {
  long i = (long)blockIdx.x * blockDim.x + threadIdx.x;
  if (i < n) p[i] = 0;
}

// ------------------------- mat-GRU weight evolution -------------------------
// Phase 1: Z = sigmoid((Wz+Uz)@Q + bz), RQ = sigmoid((Wr+Ur)@Q + br) * Q
__global__ __launch_bounds__(256) void gru_gates1(
    const float* __restrict__ Wz, const float* __restrict__ Uz, const float* __restrict__ bz,
    const float* __restrict__ Wr, const float* __restrict__ Ur, const float* __restrict__ br,
    const float* __restrict__ Q, float* __restrict__ Z, float* __restrict__ RQ) {
  __shared__ float swz[NF], swr[NF];
  const int i = blockIdx.x, j = threadIdx.x;
  swz[j] = Wz[i * NF + j] + Uz[i * NF + j];
  swr[j] = Wr[i * NF + j] + Ur[i * NF + j];
  __syncthreads();
  float az = 0.f, ar = 0.f;
  for (int k = 0; k < NF; ++k) {
    const float q = Q[k * NF + j];
    az = fmaf(swz[k], q, az);
    ar = fmaf(swr[k], q, ar);
  }
  const float z = 1.f / (1.f + expf(-(az + bz[i * NF + j])));
  const float r = 1.f / (1.f + expf(-(ar + br[i * NF + j])));
  Z[i * NF + j] = z;
  RQ[i * NF + j] = r * Q[i * NF + j];
}

// Phase 2: h = tanh(Wh@Q + Uh@RQ + bh); Qn = (1-Z)*Q + Z*h
__global__ __launch_bounds__(256) void gru_gates2(
    const float* __restrict__ Wh, const float* __restrict__ Uh, const float* __restrict__ bh,
    const float* __restrict__ Q, const float* __restrict__ Z, const float* __restrict__ RQ,
    float* __restrict__ Qn) {
  __shared__ float swh[NF], suh[NF];
  const int i = blockIdx.x, j = threadIdx.x;
  swh[j] = Wh[i * NF + j];
  suh[j] = Uh[i * NF + j];
  __syncthreads();
  float acc = 0.f;
  for (int k = 0; k < NF; ++k) {
    acc = fmaf(swh[k], Q[k * NF + j], acc);
    acc = fmaf(suh[k], RQ[k * NF + j], acc);
  }
  const float h = tanhf(acc + bh[i * NF + j]);
  const float z = Z[i * NF + j];
  Qn[i * NF + j] = (1.f - z) * Q[i * NF + j] + z * h;
}

// --------------------- fp32 WMMA GEMM: C[MxNF] = A[MxNF] @ B[NFxNF] ---------
// c_act=1 applies bias+ReLU on store.
#define BM 64
#define BN 32
#define KC 32

__global__ __launch_bounds__(256) void gemm_f32_wmma(
    const float* __restrict__ A, const float* __restrict__ B, float* __restrict__ C,
    int M, int c_act, const float* __restrict__ bias) {
  __shared__ float As[BM][KC + 1];
  __shared__ float Bs[KC][BN + 1];
  const int tid = threadIdx.x;
  const int lane = tid & 31;
  const int wid = tid >> 5;
  const int wm = wid & 3;             // 4 row tiles of 16
  const int wn = wid >> 2;            // 2 col tiles of 16
  const int l16 = lane & 15;
  const int khalf = (lane < 16) ? 0 : 2;   // ISA 32-bit A layout: K pairs per lane-half
  const int rowBase = blockIdx.x * BM;
  const int colBase = blockIdx.y * BN;

  v8f acc = {};
  for (int kb = 0; kb < NF; kb += KC) {
    // stage A tile (64x32): 512 float4, 2 per thread
    for (int i = 0; i < 2; ++i) {
      const int lin = tid + i * 256;
      const int r = lin >> 3, c4 = lin & 7;
      const int grow = rowBase + r;
      float4 v = make_float4(0.f, 0.f, 0.f, 0.f);
      if (grow < M) v = *(const float4*)(A + (size_t)grow * NF + kb + c4 * 4);
      As[r][c4 * 4 + 0] = v.x; As[r][c4 * 4 + 1] = v.y;
      As[r][c4 * 4 + 2] = v.z; As[r][c4 * 4 + 3] = v.w;
    }
    // stage B tile (32x32): 256 float4, 1 per thread
    {
      const int r = tid >> 3, c4 = tid & 7;
      const float4 v = *(const float4*)(B + (size_t)(kb + r) * NF + colBase + c4 * 4);
      Bs[r][c4 * 4 + 0] = v.x; Bs[r][c4 * 4 + 1] = v.y;
      Bs[r][c4 * 4 + 2] = v.z; Bs[r][c4 * 4 + 3] = v.w;
    }
    __syncthreads();
    for (int kk = 0; kk < KC; kk += 4) {
      v2f fa, fb;
      fa.x = As[wm * 16 + l16][kk + khalf];
      fa.y = As[wm * 16 + l16][kk + khalf + 1];
      fb.x = Bs[kk + khalf][wn * 16 + l16];
      fb.y = Bs[kk + khalf + 1][wn * 16 + l16];
      acc = __builtin_amdgcn_wmma_f32_16x16x4_f32(false, fa, false, fb, (short)0, acc,
                                                  false, false);
    }
    __syncthreads();
  }
  // epilogue: VGPR r -> M=r (lanes 0-15) / M=r+8 (lanes 16-31), N = lane%16
  const int col = colBase + wn * 16 + l16;
  const int rOff = (lane < 16) ? 0 : 8;
  const float bval = bias ? bias[col] : 0.f;
  for (int r = 0; r < 8; ++r) {
    const int row = rowBase + wm * 16 + r + rOff;
    if (row < M) {
      float v = acc[r] + bval;
      if (c_act) v = fmaxf(v, 0.f);
      C[(size_t)row * NF + col] = v;
    }
  }
}

// ------------------- CSR build: histogram / scan / scatter ------------------
__global__ void hist_kernel(const int* __restrict__ dst, int* __restrict__ cnt, int E) {
  int e = blockIdx.x * blockDim.x + threadIdx.x;
  if (e < E) atomicAdd(&cnt[dst[e]], 1);
}

// single-block exclusive scan (wave-shuffle based, 256 threads)
__global__ __launch_bounds__(256) void scan_kernel(const int* __restrict__ cnt,
                                                   int* __restrict__ rowptr, int N) {
  __shared__ int wsum[8];
  __shared__ int woff[8];
  __shared__ int carry_s;
  const int tid = threadIdx.x, lane = tid & 31, wid = tid >> 5;
  if (tid == 0) carry_s = 0;
  __syncthreads();
  for (int base = 0; base < N; base += 256) {
    const int i = base + tid;
    const int v = (i < N) ? cnt[i] : 0;
    int x = v;
    for (int off = 1; off < 32; off <<= 1) {
      const int t = __shfl_up(x, off, 32);
      if (lane >= off) x += t;
    }
    if (lane == 31) wsum[wid] = x;
    __syncthreads();
    if (tid < 8) {
      const int s = wsum[tid];
      int y = s;
      for (int off = 1; off < 8; off <<= 1) {
        const int t = __shfl_up(y, off, 8);
        if (tid >= off) y += t;
      }
      woff[tid] = y - s;            // exclusive wave offset
      if (tid == 7) wsum[0] = y;    // chunk total
    }
    __syncthreads();
    const int carry = carry_s;
    if (i < N) rowptr[i] = carry + woff[wid] + (x - v);
    const int total = wsum[0];
    __syncthreads();
    if (tid == 0) carry_s = carry + total;
    __syncthreads();
  }
  if (threadIdx.x == 0) rowptr[N] = carry_s;
}

__global__ void scatter_kernel(const int* __restrict__ src, const int* __restrict__ dst,
                               const float* __restrict__ val, const int* __restrict__ rowptr,
                               int* __restrict__ cursor, int* __restrict__ sSrc,
                               float* __restrict__ sVal, int E) {
  int e = blockIdx.x * blockDim.x + threadIdx.x;
  if (e >= E) return;
  const int d = dst[e];
  const int pos = rowptr[d] + atomicAdd(&cursor[d], 1);
  sSrc[pos] = src[e];
  sVal[pos] = val[e];
}

// ----------- CSR aggregation, wave per destination node, no atomics ---------
// out[n,:] = RReLU( sum_i val_i * XW[src_i,:] )  over edges with dst == n
__global__ __launch_bounds__(256) void agg_csr(
    const float* __restrict__ XW, const int* __restrict__ rowptr,
    const int* __restrict__ sSrc, const float* __restrict__ sVal,
    float* __restrict__ out, int N) {
  const long gid = (long)blockIdx.x * blockDim.x + threadIdx.x;
  const long n = gid >> 5;
  if (n >= N) return;
  const int lane = gid & 31;
  const int beg = rowptr[n], end = rowptr[n + 1];
  float4 a0 = make_float4(0.f, 0.f, 0.f, 0.f);
  float4 a1 = make_float4(0.f, 0.f, 0.f, 0.f);
  for (int i = beg; i < end; ++i) {
    const int s = sSrc[i];
    const float v = sVal[i];
    if (i + 1 < end)  // global_prefetch of next gathered row (lane-spread over 1KB)
      __builtin_prefetch((const char*)(XW + (size_t)sSrc[i + 1] * NF) + lane * 32, 0, 0);
    const float4* xr = (const float4*)(XW + (size_t)s * NF);
    const float4 x0 = xr[lane];
    const float4 x1 = xr[lane + 32];
    a0.x = fmaf(v, x0.x, a0.x); a0.y = fmaf(v, x0.y, a0.y);
    a0.z = fmaf(v, x0.z, a0.z); a0.w = fmaf(v, x0.w, a0.w);
    a1.x = fmaf(v, x1.x, a1.x); a1.y = fmaf(v, x1.y, a1.y);
    a1.z = fmaf(v, x1.z, a1.z); a1.w = fmaf(v, x1.w, a1.w);
  }
  a0.x = rrelu_f(a0.x); a0.y = rrelu_f(a0.y); a0.z = rrelu_f(a0.z); a0.w = rrelu_f(a0.w);
  a1.x = rrelu_f(a1.x); a1.y = rrelu_f(a1.y); a1.z = rrelu_f(a1.z); a1.w = rrelu_f(a1.w);
  float4* orow = (float4*)(out + (size_t)n * NF);
  orow[lane] = a0;
  orow[lane + 32] = a1;
}

// ---------------- 16-class head + log_softmax (one wave per node) -----------
__global__ __launch_bounds__(256) void logits_kernel(
    const float* __restrict__ hid, const float* __restrict__ W2, const float* __restrict__ b2,
    float* __restrict__ out, int N) {
  const long gid = (long)blockIdx.x * blockDim.x + threadIdx.x;
  const long n = gid >> 5;
  if (n >= N) return;
  const int lane = gid & 31;
  const int c = lane & 15;
  const int half = lane >> 4;
  const float* hrow = hid + (size_t)n * NF;
  float acc = 0.f;
  for (int k = half * 128; k < half * 128 + 128; ++k)
    acc = fmaf(hrow[k], W2[k * 16 + c], acc);
  acc += __shfl_xor(acc, 16, 32);          // combine K halves
  acc += b2[c];
  float m = acc;
  for (int mask = 1; mask < 16; mask <<= 1) m = fmaxf(m, __shfl_xor(m, mask, 32));
  float s = expf(acc - m);
  for (int mask = 1; mask < 16; mask <<= 1) s += __shfl_xor(s, mask, 32);
  if (lane < 16) out[(size_t)n * 16 + c] = acc - m - logf(s);
}

// ---------------------------------- host ------------------------------------
extern "C" void kernel_launch(void* const* d_in, const int* in_sizes, int n_in,
                              void* d_out, int out_size, void* d_ws, size_t ws_size,
                              hipStream_t stream) {
  (void)n_in; (void)out_size; (void)ws_size;
  const float* node_feats = (const float*)d_in[0];
  const float* edge_val   = (const float*)d_in[1];
  const float* L[2][10];
  for (int l = 0; l < 2; ++l)
    for (int k = 0; k < 10; ++k) L[l][k] = (const float*)d_in[2 + l * 10 + k];
  // L[l]: 0=Wz 1=Uz 2=bz 3=Wr 4=Ur 5=br 6=Wh 7=Uh 8=bh 9=Q0
  const float* mlp_W1 = (const float*)d_in[22];
  const float* mlp_b1 = (const float*)d_in[23];
  const float* mlp_W2 = (const float*)d_in[24];
  const float* mlp_b2 = (const float*)d_in[25];
  const int* edge_src = (const int*)d_in[26];
  const int* edge_dst = (const int*)d_in[27];

  const int N = in_sizes[0] / (TSTEPS * NF);
  const int E = in_sizes[1] / TSTEPS;
  const int t = TSTEPS - 1;  // only last timestep reaches the output (exact)

  // ---- workspace carve-out (256B aligned chunks) ----
  char* wp = (char*)d_ws;
  auto carve = [&](size_t bytes) -> void* {
    void* p = (void*)wp;
    wp += (bytes + 255) & ~(size_t)255;
    return p;
  };
  const size_t QSZ = (size_t)NF * NF;
  float* qP[2][2];
  for (int l = 0; l < 2; ++l)
    for (int b = 0; b < 2; ++b) qP[l][b] = (float*)carve(QSZ * 4);
  float* Zb   = (float*)carve(QSZ * 4);
  float* RQb  = (float*)carve(QSZ * 4);
  float* bigA = (float*)carve((size_t)N * NF * 4);   // XW / hid
  float* bigB = (float*)carve((size_t)N * NF * 4);   // agg (rrelu'd)
  int*   cnt    = (int*)carve((size_t)N * 4);
  int*   rowptr = (int*)carve((size_t)(N + 1) * 4);
  int*   cursor = (int*)carve((size_t)N * 4);
  int*   sSrc   = (int*)carve((size_t)E * 4);
  float* sVal   = (float*)carve((size_t)E * 4);

  // ---- evolve Q for both layers (H never enters the GRU recurrence) ----
  const float* qfin[2];
  for (int l = 0; l < 2; ++l) {
    const float* Q = L[l][9];
    for (int s = 0; s < TSTEPS; ++s) {
      gru_gates1<<<NF, NF, 0, stream>>>(L[l][0], L[l][1], L[l][2],
                                        L[l][3], L[l][4], L[l][5], Q, Zb, RQb);
      float* Qn = qP[l][s & 1];
      gru_gates2<<<NF, NF, 0, stream>>>(L[l][6], L[l][7], L[l][8], Q, Zb, RQb, Qn);
      Q = Qn;
    }
    qfin[l] = Q;
  }

  const int* src_t = edge_src + (size_t)t * E;
  const int* dst_t = edge_dst + (size_t)t * E;
  const float* val_t = edge_val + (size_t)t * E;

  // ---- build CSR by destination once (shared by both edge passes) ----
  const int nBlocksN = (N + 255) / 256;
  const int nBlocksE = (E + 255) / 256;
  zero_i32<<<nBlocksN, 256, 0, stream>>>(cnt, N);
  zero_i32<<<nBlocksN, 256, 0, stream>>>(cursor, N);
  hist_kernel<<<nBlocksE, 256, 0, stream>>>(dst_t, cnt, E);
  scan_kernel<<<1, 256, 0, stream>>>(cnt, rowptr, N);
  scatter_kernel<<<nBlocksE, 256, 0, stream>>>(src_t, dst_t, val_t, rowptr, cursor,
                                               sSrc, sVal, E);

  const dim3 gGrid((N + BM - 1) / BM, NF / BN);
  const int aBlocks = (int)(((long)N * 32 + 255) / 256);

  // layer 0: XW = nf[t] @ Q0fin ; agg(+RReLU)
  gemm_f32_wmma<<<gGrid, 256, 0, stream>>>(node_feats + (size_t)t * N * NF, qfin[0],
                                           bigA, N, 0, nullptr);
  agg_csr<<<aBlocks, 256, 0, stream>>>(bigA, rowptr, sSrc, sVal, bigB, N);

  // layer 1: XW2 = h1 @ Q1fin ; agg(+RReLU)
  gemm_f32_wmma<<<gGrid, 256, 0, stream>>>(bigB, qfin[1], bigA, N, 0, nullptr);
  agg_csr<<<aBlocks, 256, 0, stream>>>(bigA, rowptr, sSrc, sVal, bigB, N);

  // MLP: hid = relu(h2 @ W1 + b1) ; logits + log_softmax
  gemm_f32_wmma<<<gGrid, 256, 0, stream>>>(bigB, mlp_W1, bigA, N, 1, mlp_b1);
  const int lBlocks = (int)(((long)N * 32 + 255) / 256);
  logits_kernel<<<lBlocks, 256, 0, stream>>>(bigA, mlp_W2, mlp_b2, (float*)d_out, N);
}